// GemNetTForceHead_15281493639528
// MI455X (gfx1250) — compile-verified
//
#include <hip/hip_runtime.h>
#include <stdint.h>

// ---------------- Tunables ----------------
#define TILE_EDGES      128                  // edges per wave-tile
#define TILE_BYTES      (TILE_EDGES * 12)    // 1536 B of edge_vec per tile
#define TILE_FLOATS     (TILE_BYTES / 4)     // 384
#define NBUF            2                    // double-buffered LDS slots
#define WAVES_PER_BLOCK 8
#define BLOCK_THREADS   (WAVES_PER_BLOCK * 32)
#define GRID_BLOCKS     2048

// Zero the output accumulator (harness poisons d_out with 0xAA).
__global__ void gemnet_zero_kernel(float* __restrict__ out, int n) {
    int i = blockIdx.x * blockDim.x + threadIdx.x;
    if (i < n) out[i] = 0.0f;
}

__device__ __forceinline__ void atom_add_f32(float* p, float v) {
    // relaxed, agent scope, result unused -> non-returning global_atomic_add_f32
    __hip_atomic_fetch_add(p, v, __ATOMIC_RELAXED, __HIP_MEMORY_SCOPE_AGENT);
}

// Issue 3 async b128 loads staging one tile (1536 B of edge_vec) into LDS.
// INST_OFFSET is added to BOTH the LDS and global addresses (ISA 15.18.3),
// so one address pair covers all three chunks per lane. The s_wait_dscnt 0
// guarantees prior LDS reads of the target buffer have completed before the
// async engine overwrites it.
__device__ __forceinline__ void issue_tile_async(uint32_t lds_addr, uint64_t gaddr) {
    asm volatile(
        "s_wait_dscnt 0\n\t"
        "global_load_async_to_lds_b128 %0, %1, off\n\t"
        "global_load_async_to_lds_b128 %0, %1, off offset:512\n\t"
        "global_load_async_to_lds_b128 %0, %1, off offset:1024"
        :
        : "v"(lds_addr), "v"(gaddr)
        : "memory");
}

// Scatter F_st[e] * edge_vec[e][0..2] into out[edge_idx[e]][0..2].
// edge_vec (12 B/edge stream) is staged into wave-private double-buffered LDS
// slots by the CDNA5 async-to-LDS engine; F_st / edge_idx arrive as coalesced
// b128 loads (4 edges per lane). ASYNCcnt completions are in-order, so
// s_wait_asynccnt 3 == "current tile done, next tile still in flight".
__global__ __launch_bounds__(BLOCK_THREADS)
void gemnet_scatter_force_kernel(const float* __restrict__ F,
                                 const float* __restrict__ vec,
                                 const int*   __restrict__ idx,
                                 float*       __restrict__ out,
                                 long long nEdges, int nTiles) {
    __shared__ __align__(16) float smem[WAVES_PER_BLOCK * NBUF * TILE_FLOATS];

    const int lane        = threadIdx.x & 31;
    const int waveInBlock = threadIdx.x >> 5;
    const int waveGlobal  = blockIdx.x * WAVES_PER_BLOCK + waveInBlock;
    const int nWaves      = gridDim.x * WAVES_PER_BLOCK;

    float* slot0 = smem + waveInBlock * (NBUF * TILE_FLOATS);
    float* slot1 = slot0 + TILE_FLOATS;
    // addrspace(3) flat pointer: low 32 bits == LDS byte address (ISA 10.2).
    const uint32_t lds0 = (uint32_t)(uintptr_t)slot0 + 16u * (uint32_t)lane;
    const uint32_t lds1 = (uint32_t)(uintptr_t)slot1 + 16u * (uint32_t)lane;
    const uint64_t vbase = (uint64_t)(uintptr_t)vec + 16u * (uint32_t)lane;

    int t = waveGlobal;
    int buf = 0;
    if (t < nTiles) {  // prologue: stage first tile into buffer 0
        issue_tile_async(lds0, vbase + (uint64_t)t * TILE_BYTES);
    }
    for (; t < nTiles; t += nWaves) {
        const long long e0 = (long long)t * TILE_EDGES;
        const int tn = t + nWaves;

        // Prefetch next tile into the other buffer while this one drains.
        if (tn < nTiles) {
            issue_tile_async(buf ? lds0 : lds1,
                             vbase + (uint64_t)tn * TILE_BYTES);
        }

        // Coalesced direct loads: 4 edges per lane (tile bases 512 B aligned).
        // Issued before the async wait so they overlap it.
        const float4 f4 = ((const float4*)(F + e0))[lane];
        const int4   i4 = ((const int4*)(idx + e0))[lane];

        if (tn < nTiles) {
            // 6 outstanding -> wait until only the newest 3 (next tile) remain.
            asm volatile("s_wait_asynccnt 3" ::: "memory");
        } else {
            asm volatile("s_wait_asynccnt 0" ::: "memory");
        }

        const float* slot = buf ? slot1 : slot0;
        const float fv[4] = {f4.x, f4.y, f4.z, f4.w};
        const int   av[4] = {i4.x, i4.y, i4.z, i4.w};
#pragma unroll
        for (int j = 0; j < 4; ++j) {
            const int   k = 4 * lane + j;          // edge within tile
            const float f = fv[j];
            float* o = out + 3 * (long long)av[j];
            const float vx = slot[k * 3 + 0];
            const float vy = slot[k * 3 + 1];
            const float vz = slot[k * 3 + 2];
            atom_add_f32(o + 0, f * vx);
            atom_add_f32(o + 1, f * vy);
            atom_add_f32(o + 2, f * vz);
        }
        buf ^= 1;
    }

    // Scalar tail for nEdges not divisible by TILE_EDGES (empty for 6.4M).
    const long long tailStart = (long long)nTiles * TILE_EDGES;
    const long long gtid   = (long long)blockIdx.x * blockDim.x + threadIdx.x;
    const long long stride = (long long)gridDim.x * blockDim.x;
    for (long long e = tailStart + gtid; e < nEdges; e += stride) {
        const float f = F[e];
        float* o = out + 3 * (long long)idx[e];
        atom_add_f32(o + 0, f * vec[3 * e + 0]);
        atom_add_f32(o + 1, f * vec[3 * e + 1]);
        atom_add_f32(o + 2, f * vec[3 * e + 2]);
    }
}

extern "C" void kernel_launch(void* const* d_in, const int* in_sizes, int n_in,
                              void* d_out, int out_size, void* d_ws, size_t ws_size,
                              hipStream_t stream) {
    (void)n_in; (void)d_ws; (void)ws_size;
    const float* F   = (const float*)d_in[0];   // [E,1] f32
    const float* vec = (const float*)d_in[1];   // [E,3] f32
    const int*   idx = (const int*)d_in[2];     // [E]   i32
    float* out = (float*)d_out;                 // [n_atoms,3] f32

    const long long nEdges = (long long)in_sizes[2];
    const int nTiles = (int)(nEdges / TILE_EDGES);

    // 1) zero accumulator
    {
        int threads = 256;
        int blocks  = (out_size + threads - 1) / threads;
        gemnet_zero_kernel<<<blocks, threads, 0, stream>>>(out, out_size);
    }
    // 2) scatter-add
    {
        int blocks = GRID_BLOCKS;
        const int waveTiles = (nTiles + WAVES_PER_BLOCK - 1) / WAVES_PER_BLOCK;
        if (blocks > waveTiles && waveTiles > 0) blocks = waveTiles;
        if (blocks < 1) blocks = 1;
        gemnet_scatter_force_kernel<<<blocks, BLOCK_THREADS, 0, stream>>>(
            F, vec, idx, out, nEdges, nTiles);
    }
}